// ContinuousDGM_30021821399832
// MI455X (gfx1250) — compile-verified
//
#include <hip/hip_runtime.h>
#include <hip/hip_bf16.h>

// ContinuousDGM on MI455X (gfx1250, wave32, WMMA + TDM).
//
// Never materialize the 256MB adjacency (exceeds 192MB L2). Two flash-style
// passes over on-the-fly 16x16 Gram tiles via V_WMMA_F32_16X16X4_F32 (the
// fp32 matrix path on CDNA5). The 16x64 xe j-block consumed by each tile is
// staged into LDS by the Tensor Data Mover (tensor_load_to_lds), double
// buffered and overlapped with WMMA via s_wait_tensorcnt. TDM's LDS-padding
// (64 dwords -> +2 dwords) gives a 66-float row stride so the transposed
// Gram B-fragment reads are LDS bank-conflict free.

typedef __attribute__((ext_vector_type(2))) float v2f;
typedef __attribute__((ext_vector_type(8))) float v8f;
typedef __attribute__((ext_vector_type(4))) unsigned int u32x4;
typedef __attribute__((ext_vector_type(4))) int i32x4;
typedef __attribute__((ext_vector_type(8))) int i32x8;

#define NROW 8192
#define INDIM 256
#define EMB 64
#define JSPLIT 8                 // column chunks per row-strip (parallelism)
#define JCHUNK (NROW / JSPLIT)   // 1024 columns per chunk
#define NT (JCHUNK / 16)         // 64 j-tiles per chunk
#define LDSTRIDE 66              // 64 dwords + TDM pad of 2 dwords

__device__ __forceinline__ v8f wmma_f32(v2f a, v2f b, v8f c) {
    // D = A(16x4,f32) * B(4x16,f32) + C(16x16,f32)
    return __builtin_amdgcn_wmma_f32_16x16x4_f32(
        /*neg_a=*/false, a, /*neg_b=*/false, b,
        /*c_mod=*/(short)0, c, /*reuse_a=*/false, /*reuse_b=*/false);
}

// TDM: DMA one 16x64 f32 tile of xe (row-major, row stride 64 floats) into
// LDS at lds_off, padding +2 dwords after every 64 dwords (row stride 66).
// D# layout per CDNA5 ISA 8.3/8.4. Groups 2/3 zero (2D tensor, tile_dim3/4=0).
__device__ __forceinline__ void tdm_load_tile(const float* gsrc, unsigned lds_off) {
    const unsigned long long ga = (unsigned long long)gsrc;
    u32x4 g0;
    g0.x = 1u;                                            // count=1, valid D#
    g0.y = lds_off;                                       // lds_addr (bytes)
    g0.z = (unsigned)ga;                                  // global_addr[31:0]
    g0.w = (unsigned)((ga >> 32) & 0x1FFFFFFu) | (2u << 30); // addr[56:32]|type=2
    i32x8 g1;
    g1.s0 = (int)((2u << 16)     // data_size = 4B
                | (1u << 20)     // pad_enable
                | (5u << 22)     // pad_interval: 64 dwords
                | (1u << 25));   // pad_amount:   2 dwords
    g1.s1 = (int)(64u << 16);    // tensor_dim0[15:0]=64 (bits 63:48)
    g1.s2 = (int)(8192u << 16);  // tensor_dim0[31:16]=0 | tensor_dim1[15:0]=8192
    g1.s3 = (int)(64u << 16);    // tensor_dim1[31:16]=0 | tile_dim0=64
    g1.s4 = 16;                  // tile_dim1=16 | tile_dim2=0
    g1.s5 = 64;                  // tensor_dim0_stride[31:0]=64
    g1.s6 = 0;                   // stride hi | tensor_dim1_stride lo (unused, 2D)
    g1.s7 = 0;
    const i32x4 gz = {0, 0, 0, 0};
    asm volatile("tensor_load_to_lds %0, %1, %2, %3"
                 :: "s"(g0), "s"(g1), "s"(gz), "s"(gz)
                 : "memory");
}

// ---------------------------------------------------------------- utilities
__global__ void k_zero(float* __restrict__ p, int n) {
    int i = blockIdx.x * blockDim.x + threadIdx.x;
    if (i < n) p[i] = 0.0f;
}

// sq[i] = ||xe_i||^2
__global__ void k_sq(const float* __restrict__ xe, float* __restrict__ sq) {
    int i = blockIdx.x * blockDim.x + threadIdx.x;
    if (i < NROW) {
        const float4* p = (const float4*)(xe + (size_t)i * EMB);
        float t = 0.0f;
#pragma unroll
        for (int k = 0; k < EMB / 4; ++k) {
            float4 v = p[k];
            t += v.x * v.x + v.y * v.y + v.z * v.z + v.w * v.w;
        }
        sq[i] = t;
    }
}

// s[i] = 1/s[i]  (after the atomic row-sum accumulation)
__global__ void k_invert(float* __restrict__ s) {
    int i = blockIdx.x * blockDim.x + threadIdx.x;
    if (i < NROW) s[i] = 1.0f / s[i];
}

// --------------------------------------------------- pass 0: xe = x @ W + b
// One wave per 16x16 output tile; K=256 -> 64 chained f32 WMMAs.
__global__ __launch_bounds__(32) void k_embed(const float* __restrict__ x,
                                              const float* __restrict__ W,
                                              const float* __restrict__ bias,
                                              float* __restrict__ xe) {
    const int lane = threadIdx.x;
    const int g = lane >> 4;      // half-wave select (K offset +2, M offset +8)
    const int l = lane & 15;
    const int i0 = blockIdx.x * 16;
    const int j0 = blockIdx.y * 16;
    const int row = i0 + l;

    v8f c = {};
    for (int k = 0; k < INDIM; k += 4) {
        const int ka = k + 2 * g;
        v2f a, b;
        a.x = x[(size_t)row * INDIM + ka];
        a.y = x[(size_t)row * INDIM + ka + 1];
        b.x = W[(size_t)ka * EMB + j0 + l];
        b.y = W[(size_t)(ka + 1) * EMB + j0 + l];
        c = wmma_f32(a, b, c);
    }
    const float bv = bias[j0 + l];
#pragma unroll
    for (int v = 0; v < 8; ++v) {
        const int m = i0 + v + 8 * g;           // C layout: VGPR v -> row v+8g
        xe[(size_t)m * EMB + j0 + l] = c[v] + bv;
    }
}

// Gram tile from LDS-staged j-block: B[k][n] = xe[j0+n][k] = buf[n*66 + k].
__device__ __forceinline__ void gram_tile(const float* __restrict__ bufc,
                                          const v2f af[16], int g, int l, v8f& c) {
#pragma unroll
    for (int t = 0; t < 16; ++t) {
        const int kb = 4 * t + 2 * g;
        v2f b;
        b.x = bufc[l * LDSTRIDE + kb];
        b.y = bufc[l * LDSTRIDE + kb + 1];
        c = wmma_f32(af[t], b, c);
    }
}

// ------------------------------------- pass 1: s[i] = sum_j 1/(1+D(i,j))
__device__ __forceinline__ void rowsum_step(const float* __restrict__ bufc,
                                            const float* __restrict__ sq,
                                            int j0, int l, int g,
                                            const float sqr[8], const v2f af[16],
                                            float acc[8]) {
    v8f c = {};
    gram_tile(bufc, af, g, l, c);
    const float sqc = sq[j0 + l];
#pragma unroll
    for (int v = 0; v < 8; ++v) {
        float d = fmaxf(sqr[v] + sqc - 2.0f * c[v], 0.0f);
        acc[v] += 1.0f / (1.0f + d);
    }
}

__global__ __launch_bounds__(32) void k_rowsum(const float* __restrict__ xe,
                                               const float* __restrict__ sq,
                                               float* __restrict__ s) {
    __shared__ float jbuf[2][16 * LDSTRIDE];
    const int lane = threadIdx.x;
    const int g = lane >> 4;
    const int l = lane & 15;
    const int i0 = blockIdx.x * 16;
    const int row = i0 + l;

    // Strip A fragments live in registers for all 64 column tiles.
    v2f af[16];
#pragma unroll
    for (int t = 0; t < 16; ++t) {
        const int ka = 4 * t + 2 * g;
        af[t].x = xe[(size_t)row * EMB + ka];
        af[t].y = xe[(size_t)row * EMB + ka + 1];
    }
    float sqr[8];
#pragma unroll
    for (int v = 0; v < 8; ++v) sqr[v] = sq[i0 + v + 8 * g];

    const int jbeg = blockIdx.y * JCHUNK;
    const unsigned lds0 = (unsigned)(unsigned long long)&jbuf[0][0];
    const unsigned lds1 = (unsigned)(unsigned long long)&jbuf[1][0];

    float acc[8] = {};
    tdm_load_tile(xe + (size_t)jbeg * EMB, lds0);            // prologue: tile 0
    for (int jt = 0; jt < NT - 1; ++jt) {
        tdm_load_tile(xe + (size_t)(jbeg + (jt + 1) * 16) * EMB,
                      ((jt + 1) & 1) ? lds1 : lds0);
        __builtin_amdgcn_s_wait_tensorcnt(1);                // tile jt landed
        rowsum_step(jbuf[jt & 1], sq, jbeg + jt * 16, l, g, sqr, af, acc);
    }
    __builtin_amdgcn_s_wait_tensorcnt(0);
    rowsum_step(jbuf[(NT - 1) & 1], sq, jbeg + (NT - 1) * 16, l, g, sqr, af, acc);

    // Row-sum: reduce across the 16 lanes of each half (masks<16 stay inside).
#pragma unroll
    for (int v = 0; v < 8; ++v) {
        float t = acc[v];
        t += __shfl_xor(t, 8, 32);
        t += __shfl_xor(t, 4, 32);
        t += __shfl_xor(t, 2, 32);
        t += __shfl_xor(t, 1, 32);
        if (l == 0) atomicAdd(&s[i0 + v + 8 * g], t);
    }
}

// ------------- pass 2: out[i,:] += sum_j (1/(1+D(i,j)) * sinv[j]) * xe[j,:]
__device__ __forceinline__ void prop_step(const float* __restrict__ bufc,
                                          const float* __restrict__ sq,
                                          const float* __restrict__ sinv,
                                          float* __restrict__ lw,
                                          int j0, int l, int g,
                                          const float sqr[8], const v2f af[16],
                                          v8f& a0, v8f& a1, v8f& a2, v8f& a3) {
    v8f c = {};
    gram_tile(bufc, af, g, l, c);
    const float sqc = sq[j0 + l];
    const float rs = sinv[j0 + l];
    // w(i,j) = sinv[j]/(1+max(D,0)); write tile in C layout [i][j] to LDS.
#pragma unroll
    for (int v = 0; v < 8; ++v) {
        float d = fmaxf(sqr[v] + sqc - 2.0f * c[v], 0.0f);
        lw[(v + 8 * g) * 17 + l] = rs / (1.0f + d);
    }
    __syncthreads();   // single-wave block: orders DS store->load
    // Re-read w as A fragments: A[m=l][k], K=16 -> 4 k-steps.
    v2f aw[4];
#pragma unroll
    for (int t = 0; t < 4; ++t) {
        const int ka = 4 * t + 2 * g;
        aw[t].x = lw[l * 17 + ka];
        aw[t].y = lw[l * 17 + ka + 1];
    }
    __syncthreads();
    // acc[16x64] += w[16x16] @ xe_block; B comes from the SAME staged tile.
#pragma unroll
    for (int t = 0; t < 4; ++t) {
        const int kk = 4 * t + 2 * g;
        v2f b0, b1, b2, b3;
        b0.x = bufc[kk * LDSTRIDE + 0 + l];  b0.y = bufc[(kk + 1) * LDSTRIDE + 0 + l];
        b1.x = bufc[kk * LDSTRIDE + 16 + l]; b1.y = bufc[(kk + 1) * LDSTRIDE + 16 + l];
        b2.x = bufc[kk * LDSTRIDE + 32 + l]; b2.y = bufc[(kk + 1) * LDSTRIDE + 32 + l];
        b3.x = bufc[kk * LDSTRIDE + 48 + l]; b3.y = bufc[(kk + 1) * LDSTRIDE + 48 + l];
        a0 = wmma_f32(aw[t], b0, a0);
        a1 = wmma_f32(aw[t], b1, a1);
        a2 = wmma_f32(aw[t], b2, a2);
        a3 = wmma_f32(aw[t], b3, a3);
    }
}

__global__ __launch_bounds__(32) void k_prop(const float* __restrict__ xe,
                                             const float* __restrict__ sq,
                                             const float* __restrict__ sinv,
                                             float* __restrict__ out) {
    __shared__ float jbuf[2][16 * LDSTRIDE];
    __shared__ float lw[16 * 17];
    const int lane = threadIdx.x;
    const int g = lane >> 4;
    const int l = lane & 15;
    const int i0 = blockIdx.x * 16;
    const int row = i0 + l;

    v2f af[16];
#pragma unroll
    for (int t = 0; t < 16; ++t) {
        const int ka = 4 * t + 2 * g;
        af[t].x = xe[(size_t)row * EMB + ka];
        af[t].y = xe[(size_t)row * EMB + ka + 1];
    }
    float sqr[8];
#pragma unroll
    for (int v = 0; v < 8; ++v) sqr[v] = sq[i0 + v + 8 * g];

    const int jbeg = blockIdx.y * JCHUNK;
    const unsigned lds0 = (unsigned)(unsigned long long)&jbuf[0][0];
    const unsigned lds1 = (unsigned)(unsigned long long)&jbuf[1][0];

    v8f acc0 = {}, acc1 = {}, acc2 = {}, acc3 = {};
    tdm_load_tile(xe + (size_t)jbeg * EMB, lds0);            // prologue: tile 0
    for (int jt = 0; jt < NT - 1; ++jt) {
        tdm_load_tile(xe + (size_t)(jbeg + (jt + 1) * 16) * EMB,
                      ((jt + 1) & 1) ? lds1 : lds0);
        __builtin_amdgcn_s_wait_tensorcnt(1);
        prop_step(jbuf[jt & 1], sq, sinv, lw, jbeg + jt * 16, l, g, sqr, af,
                  acc0, acc1, acc2, acc3);
    }
    __builtin_amdgcn_s_wait_tensorcnt(0);
    prop_step(jbuf[(NT - 1) & 1], sq, sinv, lw, jbeg + (NT - 1) * 16, l, g, sqr, af,
              acc0, acc1, acc2, acc3);

#pragma unroll
    for (int v = 0; v < 8; ++v) {
        const int m = i0 + v + 8 * g;
        atomicAdd(&out[(size_t)m * EMB + 0 + l], acc0[v]);
        atomicAdd(&out[(size_t)m * EMB + 16 + l], acc1[v]);
        atomicAdd(&out[(size_t)m * EMB + 32 + l], acc2[v]);
        atomicAdd(&out[(size_t)m * EMB + 48 + l], acc3[v]);
    }
}

extern "C" void kernel_launch(void* const* d_in, const int* in_sizes, int n_in,
                              void* d_out, int out_size, void* d_ws, size_t ws_size,
                              hipStream_t stream) {
    const float* x = (const float*)d_in[0];    // [8192, 256]
    const float* W = (const float*)d_in[1];    // [256, 64]
    const float* b = (const float*)d_in[2];    // [64]
    float* out = (float*)d_out;                // [8192, 64]

    float* xe = (float*)d_ws;                  // 8192*64 floats (2 MB)
    float* sq = xe + (size_t)NROW * EMB;       // 8192 floats
    float* s  = sq + NROW;                     // 8192 floats (row sums -> 1/s)

    // Zero the atomic accumulation targets (d_out/d_ws are poisoned).
    k_zero<<<(NROW + 255) / 256, 256, 0, stream>>>(s, NROW);
    k_zero<<<(NROW * EMB + 255) / 256, 256, 0, stream>>>(out, NROW * EMB);

    // xe = x @ W + b
    k_embed<<<dim3(NROW / 16, EMB / 16), 32, 0, stream>>>(x, W, b, xe);
    // sq[i] = ||xe_i||^2
    k_sq<<<NROW / 256, 256, 0, stream>>>(xe, sq);
    // s[i] = sum_j 1/(1 + D(i,j)), then invert in place
    k_rowsum<<<dim3(NROW / 16, JSPLIT), 32, 0, stream>>>(xe, sq, s);
    k_invert<<<NROW / 256, 256, 0, stream>>>(s);
    // out = (A * diag(1/s)) @ xe, fused
    k_prop<<<dim3(NROW / 16, JSPLIT), 32, 0, stream>>>(xe, sq, s, out);
}